// wmc_loss_70772471104061
// MI455X (gfx1250) — compile-verified
//
#include <hip/hip_runtime.h>
#include <math.h>

// ---- problem constants (from reference) ----
static constexpr int NC = 128;          // classes per row
#define THR_F 9.2885e-30f

// ---- wave32 reduction helpers ----
__device__ __forceinline__ float wred_max(float v) {
#pragma unroll
    for (int m = 16; m; m >>= 1) v = fmaxf(v, __shfl_xor(v, m, 32));
    return v;
}
__device__ __forceinline__ float wred_sum(float v) {
#pragma unroll
    for (int m = 16; m; m >>= 1) v += __shfl_xor(v, m, 32);
    return v;
}
__device__ __forceinline__ float wred_prod(float v) {
#pragma unroll
    for (int m = 16; m; m >>= 1) v *= __shfl_xor(v, m, 32);
    return v;
}
__device__ __forceinline__ float safe_neglog(float p) {
    return (p > THR_F) ? -logf(p) : 0.0f;
}
// select component j (wave-uniform j) of a float4
__device__ __forceinline__ float sel4(float4 v, int j) {
    float r = v.x;
    r = (j == 1) ? v.y : r;
    r = (j == 2) ? v.z : r;
    r = (j == 3) ? v.w : r;
    return r;
}

__global__ __launch_bounds__(256) void wmc_loss_70772471104061_kernel(
    const float* __restrict__ s1, const int* __restrict__ y1,
    const float* __restrict__ s2, const int* __restrict__ y2,
    float* __restrict__ out, int K)
{
    const int lane = threadIdx.x & 31;
    const int wv   = threadIdx.x >> 5;
    const int wavesPerGrid = (gridDim.x * blockDim.x) >> 5;
    int row0 = blockIdx.x * (blockDim.x >> 5) + wv;

    for (int row = row0; row < K; row += wavesPerGrid) {
        const size_t base = (size_t)row * NC;
        // prefetch next iteration's row for this wave (global_prefetch_b8)
        int nrow = row + wavesPerGrid;
        if (nrow < K) {
            const size_t nb = (size_t)nrow * NC + (size_t)lane * 4;
            __builtin_prefetch(s1 + nb, 0, 3);
            __builtin_prefetch(s2 + nb, 0, 3);
            __builtin_prefetch(y1 + nb, 0, 3);
            __builtin_prefetch(y2 + nb, 0, 3);
        }

        // one wave = one row; lane covers 4 contiguous columns (B128 loads)
        float4 a1 = ((const float4*)(s1 + base))[lane];
        float4 a2 = ((const float4*)(s2 + base))[lane];
        int4   b1 = ((const int4*)(y1 + base))[lane];
        int4   b2 = ((const int4*)(y2 + base))[lane];

        // softmax(scores1) — max-subtracted, matching jax.nn.softmax
        float m1 = wred_max(fmaxf(fmaxf(a1.x, a1.y), fmaxf(a1.z, a1.w)));
        float e1x = expf(a1.x - m1), e1y = expf(a1.y - m1);
        float e1z = expf(a1.z - m1), e1w = expf(a1.w - m1);
        float inv1 = 1.0f / wred_sum(e1x + e1y + e1z + e1w);
        float4 P1 = { e1x * inv1, e1y * inv1, e1z * inv1, e1w * inv1 };

        // softmax(scores2)
        float m2 = wred_max(fmaxf(fmaxf(a2.x, a2.y), fmaxf(a2.z, a2.w)));
        float e2x = expf(a2.x - m2), e2y = expf(a2.y - m2);
        float e2z = expf(a2.z - m2), e2w = expf(a2.w - m2);
        float inv2 = 1.0f / wred_sum(e2x + e2y + e2z + e2w);
        float4 P2 = { e2x * inv2, e2y * inv2, e2z * inv2, e2w * inv2 };

        // row classification (wave-uniform) via wave32 ballots
        bool laneEq = (b1.x == b2.x) & (b1.y == b2.y) & (b1.z == b2.z) & (b1.w == b2.w);
        unsigned eqm = __builtin_amdgcn_ballot_w32(laneEq);
        unsigned g0  = __builtin_amdgcn_ballot_w32(b1.x != 0);
        unsigned g1  = __builtin_amdgcn_ballot_w32(b1.y != 0);
        unsigned g2  = __builtin_amdgcn_ballot_w32(b1.z != 0);
        unsigned g3  = __builtin_amdgcn_ballot_w32(b1.w != 0);
        bool eq = (eqm == 0xFFFFFFFFu);
        int  s  = __popc(g0) + __popc(g1) + __popc(g2) + __popc(g3);

        // first / last nonzero column of Y1 (column = 4*lane + comp)
        int idxa = 1 << 30, idxb = -1;
        if (g0) { idxa = min(idxa, ((__ffs(g0) - 1) << 2) + 0); idxb = max(idxb, ((31 - __clz(g0)) << 2) + 0); }
        if (g1) { idxa = min(idxa, ((__ffs(g1) - 1) << 2) + 1); idxb = max(idxb, ((31 - __clz(g1)) << 2) + 1); }
        if (g2) { idxa = min(idxa, ((__ffs(g2) - 1) << 2) + 2); idxb = max(idxb, ((31 - __clz(g2)) << 2) + 2); }
        if (g3) { idxa = min(idxa, ((__ffs(g3) - 1) << 2) + 3); idxb = max(idxb, ((31 - __clz(g3)) << 2) + 3); }

        float loss;
        if (eq && s == 1) {
            // equal one-hot lineage
            int ja = idxa & 3, la = idxa >> 2;
            float p1a = __shfl(sel4(P1, ja), la, 32);
            float p2a = __shfl(sel4(P2, ja), la, 32);
            loss = safe_neglog(p1a) + safe_neglog(p2a);
        } else if (eq && s == 2) {
            // equal two-hot lineage
            int ja = idxa & 3, la = idxa >> 2;
            int jb = idxb & 3, lb = idxb >> 2;
            float p1a = __shfl(sel4(P1, ja), la, 32);
            float p2a = __shfl(sel4(P2, ja), la, 32);
            float p1b = __shfl(sel4(P1, jb), lb, 32);
            float p2b = __shfl(sel4(P2, jb), lb, 32);
            float prob2 = 1.0f - (1.0f - p1a * p2b) * (1.0f - p1b * p2a);
            loss = safe_neglog(prob2);
        } else {
            // disjoint case: D(l,p) = 1 - prod(1 - l*p)  (exact: unselected terms are 1.0)
            float l1x = (float)b1.x, l1y = (float)b1.y, l1z = (float)b1.z, l1w = (float)b1.w;
            float l2x = (float)b2.x, l2y = (float)b2.y, l2z = (float)b2.z, l2w = (float)b2.w;
            float t11 = (1.0f - l1x * P1.x) * (1.0f - l1y * P1.y) * (1.0f - l1z * P1.z) * (1.0f - l1w * P1.w);
            float t21 = (1.0f - l2x * P1.x) * (1.0f - l2y * P1.y) * (1.0f - l2z * P1.z) * (1.0f - l2w * P1.w);
            float t12 = (1.0f - l1x * P2.x) * (1.0f - l1y * P2.y) * (1.0f - l1z * P2.z) * (1.0f - l1w * P2.w);
            float t22 = (1.0f - l2x * P2.x) * (1.0f - l2y * P2.y) * (1.0f - l2z * P2.z) * (1.0f - l2w * P2.w);
            float D11 = 1.0f - wred_prod(t11);
            float D21 = 1.0f - wred_prod(t21);
            float D12 = 1.0f - wred_prod(t12);
            float D22 = 1.0f - wred_prod(t22);
            float probd = 1.0f - (1.0f - D11 * D22) * (1.0f - D21 * D12);
            loss = safe_neglog(probd);
        }

        if (lane == 0) out[row] = loss;
    }
}

extern "C" void kernel_launch(void* const* d_in, const int* in_sizes, int n_in,
                              void* d_out, int out_size, void* d_ws, size_t ws_size,
                              hipStream_t stream) {
    // setup_inputs order: scores1 (f32), Y1 (i32), scores2 (f32), Y2 (i32)
    const float* s1 = (const float*)d_in[0];
    const int*   y1 = (const int*)d_in[1];
    const float* s2 = (const float*)d_in[2];
    const int*   y2 = (const int*)d_in[3];
    float* out = (float*)d_out;

    const int K = out_size;               // one loss per row
    const int wavesPerBlock = 256 / 32;   // 8 waves per 256-thread block (wave32)
    int blocks = (K + wavesPerBlock - 1) / wavesPerBlock;
    if (blocks > 8192) blocks = 8192;     // grid-stride: 2 rows/wave at K=131072
    if (blocks < 1) blocks = 1;

    wmc_loss_70772471104061_kernel<<<blocks, 256, 0, stream>>>(s1, y1, s2, y2, out, K);
}